// RewardComputer_14860586844379
// MI455X (gfx1250) — compile-verified
//
#include <hip/hip_runtime.h>
#include <stdint.h>

// ---------------------------------------------------------------------------
// RewardComputer for MI455X (gfx1250, wave32).
// Memory-bound workload (~234 MB streamed => ~10us at 23.3 TB/s). No matrix
// ops exist in this reference, so no WMMA; the CDNA5 path used is the
// asynchronous global->LDS DMA (ASYNCcnt) to stage Sudoku boards for the
// entropy kernel, converting 324B-strided per-thread access into fully
// coalesced B128 transfers.
// ---------------------------------------------------------------------------

#define LCELLS 81
#define ENT_BLOCK 128   // boards (and threads) per workgroup in entropy kernel

// ------------------------------ rewards ------------------------------------

__device__ __forceinline__ float reward_elem(float qv, int p, int l,
                                             unsigned lockb, unsigned rejb) {
    const bool correct = (p == l);
    const bool lock    = (lockb != 0u);
    const bool rej     = (rejb  != 0u);
    const bool conf    = (qv > 0.0f);
    float r = 0.0f;
    if (lock) {
        r  = correct ? (conf ? 1.5f : 0.7f) : (conf ? -2.0f : -0.5f);
        r += 0.5f * qv * (correct ? 1.0f : -1.0f);   // consistency bonus
    }
    r += rej ? (correct ? -0.3f : 0.3f) : 0.0f;
    return r;
}

__global__ __launch_bounds__(256) void rewards_kernel(
    const float* __restrict__ q,
    const int*   __restrict__ pred,
    const int*   __restrict__ lab,
    const unsigned char* __restrict__ locked,
    const unsigned char* __restrict__ rejected,
    float* __restrict__ out,
    int n)                       // n = B*81 total elements
{
    const int n4  = n >> 2;
    const int tid = blockIdx.x * blockDim.x + threadIdx.x;

    if (tid < n4) {
        // 16B vector loads; masks packed 4-per-dword.
        const float4 qv = ((const float4*)q)[tid];
        const int4   pv = ((const int4*)pred)[tid];
        const int4   lv = ((const int4*)lab)[tid];
        const unsigned lm = ((const unsigned*)locked)[tid];
        const unsigned rm = ((const unsigned*)rejected)[tid];
        float4 r;
        r.x = reward_elem(qv.x, pv.x, lv.x,  lm        & 0xffu,  rm        & 0xffu);
        r.y = reward_elem(qv.y, pv.y, lv.y, (lm >> 8)  & 0xffu, (rm >> 8)  & 0xffu);
        r.z = reward_elem(qv.z, pv.z, lv.z, (lm >> 16) & 0xffu, (rm >> 16) & 0xffu);
        r.w = reward_elem(qv.w, pv.w, lv.w, (lm >> 24) & 0xffu, (rm >> 24) & 0xffu);
        ((float4*)out)[tid] = r;
    }
    // Scalar tail (n % 4 != 0); not taken for B=131072 but kept for generality.
    const int rem = n & 3;
    if (rem && tid == n4) {
        for (int j = 0; j < rem; ++j) {
            const int i = n4 * 4 + j;
            out[i] = reward_elem(q[i], pred[i], lab[i], locked[i], rejected[i]);
        }
    }
}

// ------------------------------ entropy ------------------------------------

__global__ __launch_bounds__(ENT_BLOCK) void entropy_kernel(
    const int* __restrict__ grid,   // [B, 9, 9]
    float* __restrict__ ent,        // [B]
    int B)
{
    __shared__ int tile[ENT_BLOCK * LCELLS];   // 41,472 B of the 320 KB LDS

    const int t      = threadIdx.x;
    const int wgBase = blockIdx.x * ENT_BLOCK;
    const int* gsrc  = grid + (size_t)wgBase * LCELLS;

    if (wgBase + ENT_BLOCK <= B) {
        // --- CDNA5 async DMA stage: global -> LDS, fully coalesced. -------
        // 128*81 = 10368 dwords = 2592 x 16B chunks = 20 full b128 rounds
        // per lane + one b32 tail (128 dwords).
        // Low 32 bits of a flat LDS pointer are the LDS byte offset.
        const unsigned ldsBase = (unsigned)(size_t)(&tile[0]);
#pragma unroll
        for (int k = 0; k < 20; ++k) {
            const int chunk = t + k * ENT_BLOCK;               // 16B chunk id
            const unsigned loff  = ldsBase + (unsigned)chunk * 16u;
            const uint64_t gaddr = (uint64_t)(size_t)(gsrc + chunk * 4);
            asm volatile("global_load_async_to_lds_b128 %0, %1, off"
                         :: "v"(loff), "v"(gaddr) : "memory");
        }
        {
            const int i = 20 * ENT_BLOCK * 4 + t;              // dwords 10240..10367
            const unsigned loff  = ldsBase + (unsigned)i * 4u;
            const uint64_t gaddr = (uint64_t)(size_t)(gsrc + i);
            asm volatile("global_load_async_to_lds_b32 %0, %1, off"
                         :: "v"(loff), "v"(gaddr) : "memory");
        }
        asm volatile("s_wait_asynccnt 0x0" ::: "memory");
    } else {
        // Partial last workgroup (not hit for B=131072): plain guarded copy.
        const int valid = B - wgBase;
        for (int i = t; i < valid * LCELLS; i += ENT_BLOCK) tile[i] = gsrc[i];
    }
    __syncthreads();

    const int b = wgBase + t;
    if (b >= B) return;

    // Thread t owns board t: LDS stride 81 dwords; 81 mod 64 = 17 (odd), so
    // all 32 lanes of a wave land in distinct banks -> conflict-free reads.
    const int* g = &tile[t * LCELLS];

    unsigned row[9], col[9], box[9];
    unsigned fill[3] = {0u, 0u, 0u};          // 81 "cell filled" bits
#pragma unroll
    for (int i = 0; i < 9; ++i) { row[i] = 0u; col[i] = 0u; box[i] = 0u; }

    // Pass A: build used-value masks (fully unrolled -> arrays stay in VGPRs).
#pragma unroll
    for (int r = 0; r < 9; ++r) {
#pragma unroll
        for (int c = 0; c < 9; ++c) {
            const int idx = r * 9 + c;
            const int v   = g[idx];                       // 0 = empty, 1..9
            const unsigned bit = (v > 0) ? (1u << ((v - 1) & 31)) : 0u;
            row[r] |= bit;
            col[c] |= bit;
            box[(r / 3) * 3 + (c / 3)] |= bit;
            fill[idx >> 5] |= (unsigned)(v > 0) << (idx & 31);
        }
    }

    // Pass B: per empty cell, entropy = log2(#candidates) (0 candidates -> 0).
    float e = 0.0f;
#pragma unroll
    for (int r = 0; r < 9; ++r) {
#pragma unroll
        for (int c = 0; c < 9; ++c) {
            const int idx = r * 9 + c;
            const unsigned used =
                (row[r] | col[c] | box[(r / 3) * 3 + (c / 3)]) & 0x1FFu;
            const float ncand = (float)(9 - __popc(used));
            const float lg = __log2f(fmaxf(ncand, 1.0f));
            const bool filled = (fill[idx >> 5] >> (idx & 31)) & 1u;
            e += filled ? 0.0f : lg;
        }
    }
    ent[b] = e;
}

// ------------------------------ launcher -----------------------------------

extern "C" void kernel_launch(void* const* d_in, const int* in_sizes, int n_in,
                              void* d_out, int out_size, void* d_ws, size_t ws_size,
                              hipStream_t stream) {
    (void)n_in; (void)out_size; (void)d_ws; (void)ws_size;

    const int*           grid     = (const int*)d_in[0];            // [B,9,9]
    const float*         q        = (const float*)d_in[1];          // [B,81]
    const int*           pred     = (const int*)d_in[2];            // [B,81]
    const int*           lab      = (const int*)d_in[3];            // [B,81]
    const unsigned char* locked   = (const unsigned char*)d_in[4];  // [B,81] bool
    const unsigned char* rejected = (const unsigned char*)d_in[5];  // [B,81] bool

    const int BL = in_sizes[1];        // B * 81
    const int B  = BL / LCELLS;

    float* rewards = (float*)d_out;            // [B*81]
    float* ent     = (float*)d_out + (size_t)BL;  // [B]

    // Kernel 1: elementwise rewards, 4 elements/thread.
    {
        const int n4      = BL >> 2;
        const int threads = n4 + ((BL & 3) ? 1 : 0);
        const int blocks  = (threads + 255) / 256;
        rewards_kernel<<<blocks, 256, 0, stream>>>(q, pred, lab, locked,
                                                   rejected, rewards, BL);
    }
    // Kernel 2: per-board entropy with async global->LDS staging.
    {
        const int blocks = (B + ENT_BLOCK - 1) / ENT_BLOCK;
        entropy_kernel<<<blocks, ENT_BLOCK, 0, stream>>>(grid, ent, B);
    }
}